// QSEBlock_58402965291549
// MI455X (gfx1250) — compile-verified
//
#include <hip/hip_runtime.h>
#include <math.h>

// ---------------------------------------------------------------------------
// QSE block for MI455X (gfx1250):
//   k1: avg-pool  (bandwidth bound, b128 loads; leaves tail of x in 192MB L2)
//   k2: quantum circuit sim in LDS + FC via V_WMMA_F32_16X16X4_F32 + sigmoid
//   k3: broadcast excite scale, REVERSE block order to harvest L2-resident x,
//       nontemporal stores so the write stream doesn't evict x lines
// ---------------------------------------------------------------------------

typedef float v2f __attribute__((ext_vector_type(2)));
typedef float v4f __attribute__((ext_vector_type(4)));
typedef float v8f __attribute__((ext_vector_type(8)));

#define NQ   9
#define NST  512          // 2^NQ amplitudes
#define NB   32           // batch
#define HW4  1024         // 64*64 floats = 1024 float4
#define KPAD 12           // K=9 padded to 3 WMMA k-steps of 4

// ---------------- Kernel 1: adaptive avg pool over HxW ----------------------
// grid = NB*512 blocks (one per (b,c) plane), 256 threads each.
__global__ void qse_pool_kernel(const float* __restrict__ x,
                                float* __restrict__ y) {
    __shared__ float red[8];
    const int bc = blockIdx.x;
    const v4f* x4 = reinterpret_cast<const v4f*>(x) + (size_t)bc * HW4;
    const int t = threadIdx.x;

    float s = 0.0f;
#pragma unroll
    for (int j = 0; j < 4; ++j) {
        v4f v = x4[t + 256 * j];
        s += v.x + v.y + v.z + v.w;
    }
    // wave32 butterfly reduction
#pragma unroll
    for (int m = 16; m > 0; m >>= 1) s += __shfl_xor(s, m, 32);
    if ((t & 31) == 0) red[t >> 5] = s;
    __syncthreads();
    if (t == 0) {
        float tot = 0.0f;
#pragma unroll
        for (int w = 0; w < 8; ++w) tot += red[w];
        y[bc] = tot * (1.0f / 4096.0f);
    }
}

// ---------------- Kernel 2: quantum sim + FC(WMMA) + sigmoid ----------------
// Single block, 512 threads (16 waves). Dynamic LDS:
//   sA[32*512] | sB[32*512] | zmat[32*12] | nrm[32]
__global__ void qse_quantum_fc_kernel(const float* __restrict__ y,
                                      const float* __restrict__ qp,
                                      const float* __restrict__ fc_w,  // [512,9]
                                      const float* __restrict__ fc_b,  // [512]
                                      float* __restrict__ excite) {    // [32,512]
    extern __shared__ float smem[];
    float* sA   = smem;                 // 16384
    float* sB   = smem + NB * NST;      // 16384
    float* zmat = smem + 2 * NB * NST;  // 32*12 (K-padded, cols 9..11 stay 0)
    float* nrm  = zmat + NB * KPAD;     // 32

    const int t    = threadIdx.x;       // 0..511
    const int b    = t >> 4;            // batch 0..31
    const int sub  = t & 15;            // 16 threads per batch
    const int ibase = sub * 32;         // 32 contiguous amplitudes each

    if (t < NB * KPAD) zmat[t] = 0.0f;
    if (t < NB) nrm[t] = 0.0f;
    __syncthreads();

    // Load pooled activations as the (unnormalized) quantum state; fold the
    // amplitude normalization into the final expectation values (circuit is
    // linear => probs scale by 1/||y||^2).
    {
        float ss = 0.0f;
#pragma unroll 4
        for (int k = 0; k < 32; ++k) {
            float v = y[b * NST + ibase + k];
            sA[b * NST + ibase + k] = v;
            ss += v * v;
        }
        atomicAdd(&nrm[b], ss);
    }
    __syncthreads();

    float* src = sA;
    float* dst = sB;

    for (int l = 0; l < 2; ++l) {
        // RY(theta) on each qubit (wire 0 = MSB -> bit position 8-q)
        for (int q = 0; q < NQ; ++q) {
            const float th = qp[l * NQ + q] * 0.5f;
            const float cg = __cosf(th);
            const float sg = __sinf(th);
            const int pos = (NQ - 1) - q;
            const int mask = 1 << pos;
#pragma unroll 4
            for (int k = 0; k < 32; ++k) {
                const int i = ibase + k;
                const float v  = src[b * NST + i];
                const float pv = src[b * NST + (i ^ mask)];
                dst[b * NST + i] = ((i >> pos) & 1) ? (sg * pv + cg * v)
                                                    : (cg * v - sg * pv);
            }
            __syncthreads();
            float* tmp = src; src = dst; dst = tmp;
        }
        // CNOT chain: control q (bit 8-q), target q+1 (bit 7-q)
        for (int q = 0; q < NQ - 1; ++q) {
            const int pc = (NQ - 1) - q;
            const int pt = pc - 1;
            const int mask = 1 << pt;
#pragma unroll 4
            for (int k = 0; k < 32; ++k) {
                const int i = ibase + k;
                dst[b * NST + i] = ((i >> pc) & 1) ? src[b * NST + (i ^ mask)]
                                                   : src[b * NST + i];
            }
            __syncthreads();
            float* tmp = src; src = dst; dst = tmp;
        }
    }

    // PauliZ expectations: z[b][w] = sum_i p_i * (1 - 2*bit_{8-w}(i))
    {
        float zl[NQ];
#pragma unroll
        for (int w = 0; w < NQ; ++w) zl[w] = 0.0f;
#pragma unroll 4
        for (int k = 0; k < 32; ++k) {
            const int i = ibase + k;
            const float v = src[b * NST + i];
            const float p = v * v;
#pragma unroll
            for (int w = 0; w < NQ; ++w)
                zl[w] += ((i >> ((NQ - 1) - w)) & 1) ? -p : p;
        }
#pragma unroll
        for (int w = 0; w < NQ; ++w) atomicAdd(&zmat[b * KPAD + w], zl[w]);
    }
    __syncthreads();

    if (t < NB * NQ) {
        const int bb = t / NQ, w = t % NQ;
        zmat[bb * KPAD + w] *= __builtin_amdgcn_rcpf(nrm[bb]);  // v_rcp_f32
    }
    __syncthreads();

    // ---- FC: [32,9] @ [9,512] via V_WMMA_F32_16X16X4_F32, K padded to 12 ---
    // 64 output tiles (2 M-tiles x 32 N-tiles), 16 waves x 4 tiles, 3 k-steps.
    const int wv = t >> 5;
    const int lane = t & 31;
    const int hi = (lane < 16) ? 0 : 1;   // lane half-group
    const int lrow = lane & 15;

    for (int j = 0; j < 4; ++j) {
        const int tile  = wv * 4 + j;
        const int mtile = tile >> 5;      // 0..1
        const int ntile = tile & 31;      // 0..31
        const int m = mtile * 16 + lrow;
        const int n = ntile * 16 + lrow;

        v8f acc = {};
#pragma unroll
        for (int kb = 0; kb < KPAD; kb += 4) {
            const int ka = kb + (hi ? 2 : 0);
            // A fragment (16x4 f32): lanes 0-15 -> K={ka,ka+1}; zmat K-padded.
            v2f afrag;
            afrag.x = zmat[m * KPAD + ka];
            afrag.y = zmat[m * KPAD + ka + 1];
            // B fragment (4x16 f32): B[k][n] = fc_w[n][k], zero beyond K=9.
            v2f bfrag;
            bfrag.x = (ka     < NQ) ? fc_w[n * NQ + ka]     : 0.0f;
            bfrag.y = (ka + 1 < NQ) ? fc_w[n * NQ + ka + 1] : 0.0f;
            acc = __builtin_amdgcn_wmma_f32_16x16x4_f32(
                /*neg_a=*/false, afrag, /*neg_b=*/false, bfrag,
                /*c_mod=*/(short)0, acc, /*reuse_a=*/false, /*reuse_b=*/false);
        }
        const float bias = fc_b[n];
#pragma unroll
        for (int r = 0; r < 8; ++r) {
            const int mr = mtile * 16 + r + (hi ? 8 : 0);
            const float v = acc[r] + bias;
            // sigmoid with v_rcp_f32 (avoid IEEE-div Newton chains)
            excite[mr * NST + n] = __builtin_amdgcn_rcpf(1.0f + __expf(-v));
        }
    }
}

// ---------------- Kernel 3: out = x * excite[b,c] (broadcast) ---------------
// Reverse block order: consume the L2-resident tail of x (left there by the
// pool kernel) first. Threads within a block stay ascending -> coalesced b128.
// Nontemporal stores keep the 268MB write stream from evicting resident x.
__global__ void qse_scale_kernel(const float* __restrict__ x,
                                 const float* __restrict__ excite,
                                 float* __restrict__ out) {
    const unsigned rb = (gridDim.x - 1u) - blockIdx.x;        // reversed
    const size_t idx = (size_t)rb * blockDim.x + threadIdx.x; // float4 idx
    const v4f* x4 = reinterpret_cast<const v4f*>(x);
    v4f* o4 = reinterpret_cast<v4f*>(out);
    const float e = excite[idx >> 10];          // 1024 float4 per (b,c) plane
    v4f v = x4[idx];
    v *= e;
    __builtin_nontemporal_store(v, &o4[idx]);
}

// ---------------------------------------------------------------------------
extern "C" void kernel_launch(void* const* d_in, const int* in_sizes, int n_in,
                              void* d_out, int out_size, void* d_ws, size_t ws_size,
                              hipStream_t stream) {
    const float* x    = (const float*)d_in[0];  // [32,512,64,64]
    const float* qp   = (const float*)d_in[1];  // [2,9]
    const float* fc_w = (const float*)d_in[2];  // [512,9]
    const float* fc_b = (const float*)d_in[3];  // [512]
    float* out = (float*)d_out;

    float* y      = (float*)d_ws;   // [32*512] pooled
    float* excite = y + NB * NST;   // [32*512] gate values

    // 1) pool: one block per (b,c) plane
    qse_pool_kernel<<<NB * NST, 256, 0, stream>>>(x, y);

    // 2) quantum sim + WMMA FC: single block, 512 threads
    const size_t shmem = (size_t)(2 * NB * NST + NB * KPAD + NB) * sizeof(float);
    qse_quantum_fc_kernel<<<1, 512, shmem, stream>>>(y, qp, fc_w, fc_b, excite);

    // 3) scale: 16.7M float4 elements, reversed block order
    const int total4 = NB * NST * HW4;          // 16,777,216
    qse_scale_kernel<<<total4 / 256, 256, 0, stream>>>(x, excite, out);
}